// PixelMixBlock_54949811585180
// MI455X (gfx1250) — compile-verified
//
#include <hip/hip_runtime.h>
#include <hip/hip_bf16.h>
#include <math.h>

#define N_B   64
#define C_IN  256
#define C_INT 128
#define CP    64       // C_INT / 2  (channel pairs)
#define H_    28
#define W_    28
#define HW    784      // 49 * 16
#define SF    8
#define OH    224
#define OW    224
#define NTILE_P 49     // HW / 16
#define NTILE_M 8      // C_INT / 16
#define NQPAIR  25     // ceil(NTILE_P / 2)

typedef __attribute__((ext_vector_type(2))) float v2f;
typedef __attribute__((ext_vector_type(4))) float v4f;
typedef __attribute__((ext_vector_type(8))) float v8f;

// Q is stored channel-pair interleaved:  Qp[n][c>>1][p][c&1]
// so that the two channels a lane feeds into one WMMA K-step are one b64 load.
__device__ __forceinline__ size_t qp_pair_off(int cp, int p) {
  return ((size_t)cp * HW + p) * 2;
}

// ---------------------------------------------------------------------------
// Kernel 1: Q projection.  Q[n] = Wq[128,256] x X[n][256,784] + bq,
// written in the pair-interleaved layout. One wave per 16x16 tile, fp32 WMMA.
// ---------------------------------------------------------------------------
__global__ __launch_bounds__(32) void proj_q_kernel(const float* __restrict__ x,
                                                    const float* __restrict__ Wq,
                                                    const float* __restrict__ bq,
                                                    float* __restrict__ Qp) {
  const int tile = blockIdx.x;
  const int n  = tile / (NTILE_M * NTILE_P);
  const int r  = tile % (NTILE_M * NTILE_P);
  const int mt = r / NTILE_P;
  const int pt = r % NTILE_P;
  const int lane = threadIdx.x & 31;
  const int half = lane >> 4;
  const int l16  = lane & 15;

  const float* xn   = x  + (size_t)n * C_IN * HW;
  const float* wrow = Wq + (size_t)(mt * 16 + l16) * C_IN;
  const int pcol = pt * 16 + l16;

  v8f acc = {0.f, 0.f, 0.f, 0.f, 0.f, 0.f, 0.f, 0.f};
#pragma unroll 8
  for (int c0 = 0; c0 < C_IN; c0 += 4) {
    const int ca = c0 + 2 * half;
    v2f a = *(const v2f*)(wrow + ca);                       // contiguous in Wq
    v2f b; b.x = xn[(size_t)ca * HW + pcol];
           b.y = xn[(size_t)(ca + 1) * HW + pcol];
    acc = __builtin_amdgcn_wmma_f32_16x16x4_f32(false, a, false, b,
                                                (short)0, acc, false, false);
  }
  float* qn = Qp + (size_t)n * CP * HW * 2;
#pragma unroll
  for (int j = 0; j < 8; j += 2) {
    const int m  = mt * 16 + j + 8 * half;   // even row of the pair
    v2f o; o.x = acc[j] + bq[m];  o.y = acc[j + 1] + bq[m + 1];
    *(v2f*)(qn + qp_pair_off(m >> 1, pcol)) = o;            // b64 store
  }
}

// ---------------------------------------------------------------------------
// Kernel 1b: value projection. Vall[n][784] = Wv[1,256] . X[n] + bv  (tiny)
// ---------------------------------------------------------------------------
__global__ void value_kernel(const float* __restrict__ x,
                             const float* __restrict__ Wv,
                             const float* __restrict__ bv,
                             float* __restrict__ Vall) {
  const int t = blockIdx.x * blockDim.x + threadIdx.x;
  if (t >= N_B * HW) return;
  const int n = t / HW, p = t % HW;
  const float* xn = x + (size_t)n * C_IN * HW + p;
  float s = 0.f;
#pragma unroll 4
  for (int c = 0; c < C_IN; ++c) s = fmaf(Wv[c], xn[(size_t)c * HW], s);
  Vall[t] = s + bv[0];
}

// ---------------------------------------------------------------------------
// Kernel 2: fused attention, transposed score tiles, TWO q-tiles per wave.
//   A = K-pixel tile (one b64 load per K-step), reused by both q-tiles'
//   WMMAs -> 32 loads : 64 wmma per k-tile iteration.
//   D[m][n] = score[k][q]: each q-column is per-lane across the accumulator,
//   so the online softmax is pure per-lane scalar state (no shuffles in loop).
//   The 784x784 score matrix is never materialized (saves 157 MB of HBM/L2
//   traffic; Q (25.7 MB) stays resident in the 192 MB L2).
// ---------------------------------------------------------------------------
__global__ __launch_bounds__(32) void attn_kernel(const float* __restrict__ Qp,
                                                  const float* __restrict__ Vall,
                                                  const int* __restrict__ index,
                                                  float* __restrict__ maskws) {
  const int blk = blockIdx.x;
  const int n   = blk / NQPAIR;
  const int qp  = blk % NQPAIR;
  const int qt0 = 2 * qp;
  const bool two = (qt0 + 1) < NTILE_P;
  const int qt1 = two ? qt0 + 1 : qt0;      // wave-uniform duplicate for tail
  const int lane = threadIdx.x & 31;
  const int half = lane >> 4;
  const int l16  = lane & 15;
  const int idx  = index[n];

  const float* Qn = Qp + (size_t)n   * CP * HW * 2;   // queries (B operands)
  const float* Qk = Qp + (size_t)idx * CP * HW * 2;   // keys    (A operands)
  const float* vbase = Vall + (size_t)idx * HW;

  const int qcol0 = qt0 * 16 + l16;
  const int qcol1 = qt1 * 16 + l16;

  // Preload fixed B operands for both q-tiles: 2 x 32 x b64 = 128 VGPRs.
  v2f breg0[32], breg1[32];
#pragma unroll
  for (int s = 0; s < 32; ++s) {
    const int cp = 2 * s + half;            // channels {4s+2h, 4s+2h+1}
    breg0[s] = *(const v2f*)(Qn + qp_pair_off(cp, qcol0));
    breg1[s] = *(const v2f*)(Qn + qp_pair_off(cp, qcol1));
  }

  const float scale = 0.0883883476483184f;  // 1/sqrt(128)
  float lmax0 = -1e30f, lsum0 = 0.f, lwv0 = 0.f;
  float lmax1 = -1e30f, lsum1 = 0.f, lwv1 = 0.f;

  for (int kt = 0; kt < NTILE_P; ++kt) {
    const int k0 = kt * 16;
    v8f acc0 = {0.f, 0.f, 0.f, 0.f, 0.f, 0.f, 0.f, 0.f};
    v8f acc1 = {0.f, 0.f, 0.f, 0.f, 0.f, 0.f, 0.f, 0.f};
#pragma unroll
    for (int s = 0; s < 32; ++s) {
      const int cp = 2 * s + half;
      const v2f a = *(const v2f*)(Qk + qp_pair_off(cp, k0 + l16));  // one b64
      acc0 = __builtin_amdgcn_wmma_f32_16x16x4_f32(false, a, false, breg0[s],
                                                   (short)0, acc0, false, false);
      acc1 = __builtin_amdgcn_wmma_f32_16x16x4_f32(false, a, false, breg1[s],
                                                   (short)0, acc1, false, false);
    }
    // v for rows k0+8*half .. +7 : two aligned float4 loads, shared by both tiles.
    const v4f va = *(const v4f*)(vbase + k0 + 8 * half);
    const v4f vc = *(const v4f*)(vbase + k0 + 8 * half + 4);
#pragma unroll
    for (int j = 0; j < 8; ++j) {
      const float vk = (j < 4) ? va[j] : vc[j - 4];
      {
        const float sv = acc0[j] * scale;
        const float nm   = fmaxf(lmax0, sv);
        const float corr = __expf(lmax0 - nm);
        const float e    = __expf(sv - nm);
        lsum0 = lsum0 * corr + e;
        lwv0  = lwv0  * corr + e * vk;
        lmax0 = nm;
      }
      {
        const float sv = acc1[j] * scale;
        const float nm   = fmaxf(lmax1, sv);
        const float corr = __expf(lmax1 - nm);
        const float e    = __expf(sv - nm);
        lsum1 = lsum1 * corr + e;
        lwv1  = lwv1  * corr + e * vk;
        lmax1 = nm;
      }
    }
  }
  // Merge lane halves (l <-> l+16) per q-column, then store.
  {
    const float om = __shfl_xor(lmax0, 16, 32);
    const float os = __shfl_xor(lsum0, 16, 32);
    const float ow = __shfl_xor(lwv0,  16, 32);
    const float m2 = fmaxf(lmax0, om);
    const float fa = __expf(lmax0 - m2), fb = __expf(om - m2);
    const float s2 = lsum0 * fa + os * fb;
    const float w2 = lwv0  * fa + ow * fb;
    if (lane < 16) maskws[(size_t)n * HW + qcol0] = w2 / s2;
  }
  {
    const float om = __shfl_xor(lmax1, 16, 32);
    const float os = __shfl_xor(lsum1, 16, 32);
    const float ow = __shfl_xor(lwv1,  16, 32);
    const float m2 = fmaxf(lmax1, om);
    const float fa = __expf(lmax1 - m2), fb = __expf(om - m2);
    const float s2 = lsum1 * fa + os * fb;
    const float w2 = lwv1  * fa + ow * fb;
    if (two && lane < 16) maskws[(size_t)n * HW + qcol1] = w2 / s2;
  }
}

// ---------------------------------------------------------------------------
// Kernel 3: bilinear x8 upsample (half-pixel, align_corners=False), sigmoid,
// write [1-m, m] channels.
// ---------------------------------------------------------------------------
__global__ void upsample_kernel(const float* __restrict__ maskws,
                                float* __restrict__ out) {
  const int t = blockIdx.x * blockDim.x + threadIdx.x;
  if (t >= N_B * OH * OW) return;
  const int n = t / (OH * OW);
  const int r = t % (OH * OW);
  const int oy = r / OW, ox = r % OW;

  const float sy = (oy + 0.5f) * (1.0f / SF) - 0.5f;
  const float sx = (ox + 0.5f) * (1.0f / SF) - 0.5f;
  const int y0 = (int)floorf(sy), x0 = (int)floorf(sx);
  const float fy = sy - (float)y0, fx = sx - (float)x0;
  const int y0c = y0     < 0 ? 0 : (y0     > H_ - 1 ? H_ - 1 : y0);
  const int y1c = y0 + 1 < 0 ? 0 : (y0 + 1 > H_ - 1 ? H_ - 1 : y0 + 1);
  const int x0c = x0     < 0 ? 0 : (x0     > W_ - 1 ? W_ - 1 : x0);
  const int x1c = x0 + 1 < 0 ? 0 : (x0 + 1 > W_ - 1 ? W_ - 1 : x0 + 1);

  const float* mn = maskws + (size_t)n * HW;
  const float v00 = mn[y0c * W_ + x0c], v01 = mn[y0c * W_ + x1c];
  const float v10 = mn[y1c * W_ + x0c], v11 = mn[y1c * W_ + x1c];
  const float v = (v00 * (1.f - fx) + v01 * fx) * (1.f - fy)
                + (v10 * (1.f - fx) + v11 * fx) * fy;
  const float sig = 1.0f / (1.0f + __expf(-v));
  out[((size_t)n * 2 + 0) * (OH * OW) + r] = 1.0f - sig;
  out[((size_t)n * 2 + 1) * (OH * OW) + r] = sig;
}

// ---------------------------------------------------------------------------
extern "C" void kernel_launch(void* const* d_in, const int* in_sizes, int n_in,
                              void* d_out, int out_size, void* d_ws, size_t ws_size,
                              hipStream_t stream) {
  const float* x   = (const float*)d_in[0];
  // d_in[1] = lam (unused by the reference computation)
  const int*   idx = (const int*)d_in[2];
  const float* Wq  = (const float*)d_in[3];
  const float* bq  = (const float*)d_in[4];
  const float* Wv  = (const float*)d_in[5];
  const float* bv  = (const float*)d_in[6];
  // d_in[7] = scale_factor (== 8, fixed by the reference)
  float* out = (float*)d_out;

  float* Qp   = (float*)d_ws;                       // N * 128 * 784 floats (25.7 MB)
  float* Vall = Qp + (size_t)N_B * C_INT * HW;      // N * 784
  float* mws  = Vall + (size_t)N_B * HW;            // N * 784

  proj_q_kernel<<<N_B * NTILE_M * NTILE_P, 32, 0, stream>>>(x, Wq, bq, Qp);
  value_kernel<<<(N_B * HW + 255) / 256, 256, 0, stream>>>(x, Wv, bv, Vall);
  attn_kernel<<<N_B * NQPAIR, 32, 0, stream>>>(Qp, Vall, idx, mws);
  upsample_kernel<<<(N_B * OH * OW + 255) / 256, 256, 0, stream>>>(mws, out);
}